// ScorePredictor_15625091022922
// MI455X (gfx1250) — compile-verified
//
#include <hip/hip_runtime.h>
#include <hip/hip_bf16.h>

// u_dot_v edge scoring: score[e] = dot(x[src[e]], x[dst[e]]), D=64, f32.
// Gather-bound: x table (25.6 MB) is L2-resident on MI455X (192 MB L2);
// compute via V_WMMA_F32_16X16X4_F32, 16 edges per wave32, diag extraction.

typedef __attribute__((ext_vector_type(2))) float v2f;
typedef __attribute__((ext_vector_type(8))) float v8f;

__global__ void __launch_bounds__(256) edge_dot_wmma(
    const float* __restrict__ x,
    const long long* __restrict__ src,
    const long long* __restrict__ dst,
    float* __restrict__ out,
    int n_edges, int n_tiles)
{
    const int lane          = threadIdx.x & 31;
    const int waveInBlock   = threadIdx.x >> 5;
    const int wavesPerBlock = blockDim.x >> 5;
    const int waveGlobal    = blockIdx.x * wavesPerBlock + waveInBlock;
    const int waveCount     = gridDim.x * wavesPerBlock;

    const int m    = lane & 15;   // edge-in-tile owned by this lane (A row / B col)
    const int half = lane >> 4;   // 0 -> K pair {4t,4t+1}; 1 -> K pair {4t+2,4t+3}

    for (int tile = waveGlobal; tile < n_tiles; tile += waveCount) {
        const int ebase = tile * 16;
        int e  = ebase + m;
        int ec = (e < n_edges) ? e : (n_edges - 1);   // clamp: keep EXEC all-1s for WMMA

        // Row base pointers for this lane's edge (A side = src, B side = dst).
        const float* ps = x + (size_t)src[ec] * 64;
        const float* pd = x + (size_t)dst[ec] * 64;

        v8f c = {0.f, 0.f, 0.f, 0.f, 0.f, 0.f, 0.f, 0.f};

        // K = 64 in 16 slices of 4. A: 16x4 f32 (lane m holds row m, VGPR0/1 = K pair);
        // B: 4x16 f32 (lane m holds col m, same K-pair striping). Identical addressing.
        #pragma unroll
        for (int t = 0; t < 16; ++t) {
            v2f a = *(const v2f*)(ps + 4 * t + 2 * half);
            v2f b = *(const v2f*)(pd + 4 * t + 2 * half);
            // 8 args: (neg_a, A, neg_b, B, c_mod, C, reuse_a, reuse_b)
            c = __builtin_amdgcn_wmma_f32_16x16x4_f32(
                    false, a, false, b, (short)0, c, false, false);
        }

        // D (16x16 f32) layout: lanes 0-15: N=lane, VGPR r -> M=r.
        //                       lanes 16-31: N=lane-16, VGPR r -> M=r+8.
        // Diagonal (e,e): e<8  -> lane e,     c[e]
        //                 e>=8 -> lane e+16,  c[e-8]
        int sel = -1;
        int eo  = 0;
        if (lane < 8) {
            sel = lane;
            eo  = ebase + lane;
        } else if (lane >= 24) {
            sel = lane - 24;
            eo  = ebase + (lane - 16);
        }
        if (sel >= 0) {
            float v = c[0];
            #pragma unroll
            for (int i = 1; i < 8; ++i)
                if (sel == i) v = c[i];
            if (eo < n_edges) out[eo] = v;
        }
    }
}

extern "C" void kernel_launch(void* const* d_in, const int* in_sizes, int n_in,
                              void* d_out, int out_size, void* d_ws, size_t ws_size,
                              hipStream_t stream) {
    const float*     x   = (const float*)d_in[0];
    const long long* src = (const long long*)d_in[1];
    const long long* dst = (const long long*)d_in[2];
    float*           out = (float*)d_out;

    const int n_edges = in_sizes[1];
    const int n_tiles = (n_edges + 15) / 16;

    const int threads        = 256;               // 8 waves per block
    const int wavesPerBlock  = threads / 32;
    int blocks = (n_tiles + wavesPerBlock - 1) / wavesPerBlock;
    if (blocks > 8192) blocks = 8192;             // persistent grid-stride over tiles
    if (blocks < 1) blocks = 1;

    edge_dot_wmma<<<blocks, threads, 0, stream>>>(x, src, dst, out, n_edges, n_tiles);
}